// HeteroGNN_80530636800601
// MI455X (gfx1250) — compile-verified
//
#include <hip/hip_runtime.h>
#include <hip/hip_bf16.h>

// ---------------------------------------------------------------------------
// HeteroGNN (GAT-style) for MI455X / gfx1250, wave32.
//   1) x_l = x_src@W_l+b_l, x_r = x_dst@W_r+b_r        (f32 WMMA GEMM)
//   2) per-edge logits with fused edge_attr@W_e         (VALU, BW-bound)
//   3) CSR build by counting sort (int atomics only)
//   4) wave-per-dst segment softmax + gather-accumulate (NO float atomics)
//   5) MLP: relu(out@W1+b1)@W2+b2                       (f32 WMMA GEMM)
// ---------------------------------------------------------------------------

typedef __attribute__((ext_vector_type(2))) float v2f;
typedef __attribute__((ext_vector_type(8))) float v8f;

#define NEG_SLOPE 0.2f

// ---------------------------------------------------------------------------
// GEMM  C[M x 128] = act(A[M x 128] @ B[128 x 128] + bias)
// Block = 256 threads = 8 waves; block owns a 16-row strip (staged in LDS,
// pad-132 stride so the 16-lane column reads are bank-conflict-free); wave w
// owns columns [16w,16w+16).  K-loop: 32 x V_WMMA_F32_16X16X4_F32.
// All M values here (100000 / 50000) are multiples of 16 -> EXEC all-ones.
// ---------------------------------------------------------------------------
template <int ACT>  // 0 = none, 1 = relu
__global__ __launch_bounds__(256) void gemm128_wmma(
    const float* __restrict__ A, const float* __restrict__ B,
    const float* __restrict__ bias, float* __restrict__ C, int M) {
    __shared__ float As[16][132];

    const int row0 = blockIdx.x << 4;
    for (int i = threadIdx.x; i < 16 * 128; i += 256) {
        int r = i >> 7, c = i & 127;
        As[r][c] = A[(long)(row0 + r) * 128 + c];
    }
    __syncthreads();

    const int lane = threadIdx.x & 31;
    const int wave = threadIdx.x >> 5;
    const int m    = lane & 15;          // A row for this lane
    const int koff = (lane >> 4) << 1;   // lanes 16-31 carry K+2,K+3
    const int n    = (wave << 4) + (lane & 15);

    v8f acc = {};
#pragma unroll 4
    for (int k = 0; k < 128; k += 4) {
        v2f a, b;
        a.x = As[m][k + koff];
        a.y = As[m][k + koff + 1];
        b.x = B[(k + koff) * 128 + n];
        b.y = B[(k + koff + 1) * 128 + n];
        acc = __builtin_amdgcn_wmma_f32_16x16x4_f32(
            false, a, false, b, (short)0, acc, false, false);
    }

    const float bv = bias[n];
#pragma unroll
    for (int i = 0; i < 8; ++i) {
        int rm  = i + ((lane >> 4) << 3);  // C/D: M = i + 8*(lane/16)
        float v = acc[i] + bv;
        if (ACT) v = v > 0.f ? v : 0.f;
        C[(long)(row0 + rm) * 128 + n] = v;
    }
}

// ---------------------------------------------------------------------------
// CSR build, step 0: zero degree + cursor counters
// ---------------------------------------------------------------------------
__global__ __launch_bounds__(256) void zero_counts_kernel(
    int* __restrict__ deg, int* __restrict__ cursor, int ndst) {
    int i = blockIdx.x * blockDim.x + threadIdx.x;
    if (i < ndst) { deg[i] = 0; cursor[i] = 0; }
}

// CSR step 1: histogram of dst indices
__global__ __launch_bounds__(256) void hist_kernel(
    const int* __restrict__ dst, int* __restrict__ deg, int E) {
    int e = blockIdx.x * blockDim.x + threadIdx.x;
    if (e < E) atomicAdd(&deg[dst[e]], 1);
}

// CSR step 2: exclusive scan (single block, 1024 threads, chunked + LDS scan)
__global__ __launch_bounds__(1024) void exscan_kernel(
    const int* __restrict__ deg, int* __restrict__ off, int n) {
    __shared__ int sums[1024];
    const int t   = threadIdx.x;
    const int per = (n + 1023) / 1024;
    const int s0  = t * per;
    const int s1  = min(s0 + per, n);

    int s = 0;
    for (int i = s0; i < s1; ++i) s += deg[i];
    sums[t] = s;
    __syncthreads();
    // Hillis-Steele inclusive scan over the 1024 chunk sums
    for (int d = 1; d < 1024; d <<= 1) {
        int v = (t >= d) ? sums[t - d] : 0;
        __syncthreads();
        if (t >= d) sums[t] += v;
        __syncthreads();
    }
    int excl = (t == 0) ? 0 : sums[t - 1];
    for (int i = s0; i < s1; ++i) { off[i] = excl; excl += deg[i]; }
    if (s0 < n && s1 == n) off[n] = excl;  // total
}

// CSR step 3: scatter edge ids into segment order
__global__ __launch_bounds__(256) void scatter_kernel(
    const int* __restrict__ dst, const int* __restrict__ off,
    int* __restrict__ cursor, int* __restrict__ eidx, int E) {
    int e = blockIdx.x * blockDim.x + threadIdx.x;
    if (e >= E) return;
    int d   = dst[e];
    int pos = off[d] + atomicAdd(&cursor[d], 1);
    eidx[pos] = e;
}

// ---------------------------------------------------------------------------
// Edge pass: logit[e] = sum_c( lrelu(x_l[s,c]+x_r[d,c]+e_c) * att_c )
// e_c = edge_attr[e,:16] @ W_e[:,c] + b_e[c]  (fused, never materialized).
// One wave per edge; lane owns 4 contiguous channels -> one coalesced
// global_load_b128 per endpoint row.  W_e / att / b_e staged in LDS.
// ---------------------------------------------------------------------------
__global__ __launch_bounds__(256) void edge_logits_kernel(
    const float* __restrict__ xl, const float* __restrict__ xr,
    const float* __restrict__ eattr, const float* __restrict__ We,
    const float* __restrict__ be, const float* __restrict__ att,
    const int* __restrict__ src, const int* __restrict__ dst,
    float* __restrict__ logits, int E) {
    __shared__ float sWe[16 * 128];
    __shared__ float sAtt[128];
    __shared__ float sBe[128];
    for (int i = threadIdx.x; i < 16 * 128; i += 256) sWe[i] = We[i];
    if (threadIdx.x < 128) {
        sAtt[threadIdx.x] = att[threadIdx.x];
        sBe[threadIdx.x]  = be[threadIdx.x];
    }
    __syncthreads();

    const int lane = threadIdx.x & 31;
    const int e    = (blockIdx.x << 3) + (threadIdx.x >> 5);
    if (e >= E) return;

    const int s = src[e];
    const int d = dst[e];

    // wave-uniform 64B attr row (4 x b128)
    const float4* __restrict__ ar4 = (const float4*)(eattr + (long)e * 16);
    float av[16];
#pragma unroll
    for (int q = 0; q < 4; ++q) {
        float4 v = ar4[q];
        av[4 * q + 0] = v.x; av[4 * q + 1] = v.y;
        av[4 * q + 2] = v.z; av[4 * q + 3] = v.w;
    }

    const float4 xj = ((const float4*)(xl + (long)s * 128))[lane];
    const float4 xi = ((const float4*)(xr + (long)d * 128))[lane];
    const float xjv[4] = {xj.x, xj.y, xj.z, xj.w};
    const float xiv[4] = {xi.x, xi.y, xi.z, xi.w};

    const int c0 = lane << 2;
    float partial = 0.f;
#pragma unroll
    for (int jj = 0; jj < 4; ++jj) {
        const int c = c0 + jj;
        float ev = sBe[c];
#pragma unroll
        for (int k = 0; k < 16; ++k) ev = fmaf(av[k], sWe[k * 128 + c], ev);
        float z = xjv[jj] + xiv[jj] + ev;
        z = (z > 0.f) ? z : NEG_SLOPE * z;
        partial = fmaf(z, sAtt[c], partial);
    }
#pragma unroll
    for (int o = 16; o > 0; o >>= 1) partial += __shfl_xor(partial, o, 32);

    if (lane == 0) logits[e] = partial;
}

// ---------------------------------------------------------------------------
// Segment softmax + message accumulation, one wave per destination.
// No float atomics: max / exp-sum are wave reductions over the segment's
// edges; accumulation walks the segment with wave-uniform edge scalars and
// per-lane float4 channel ownership.  Residual x_r is the accumulator init.
// ---------------------------------------------------------------------------
__global__ __launch_bounds__(256) void dst_reduce_kernel(
    const float* __restrict__ xl, const float* __restrict__ xr,
    const float* __restrict__ logits, const float* __restrict__ ew,
    const int* __restrict__ off, const int* __restrict__ eidx,
    const int* __restrict__ src, float* __restrict__ outp, int ndst) {
    const int lane = threadIdx.x & 31;
    const int d    = (blockIdx.x << 3) + (threadIdx.x >> 5);
    if (d >= ndst) return;

    const int o0  = off[d];
    const int o1  = off[d + 1];
    const int deg = o1 - o0;

    float4 acc = ((const float4*)(xr + (long)d * 128))[lane];  // residual

    if (deg > 0) {
        // segment max
        float mx = -INFINITY;
        for (int i = lane; i < deg; i += 32)
            mx = fmaxf(mx, logits[eidx[o0 + i]]);
#pragma unroll
        for (int o = 16; o > 0; o >>= 1)
            mx = fmaxf(mx, __shfl_xor(mx, o, 32));
        // segment sum of exp
        float sm = 0.f;
        for (int i = lane; i < deg; i += 32)
            sm += __expf(logits[eidx[o0 + i]] - mx);
#pragma unroll
        for (int o = 16; o > 0; o >>= 1) sm += __shfl_xor(sm, o, 32);
        const float inv = 1.f / sm;

        // weighted gather-accumulate (wave-uniform per-edge scalar)
        for (int i = 0; i < deg; ++i) {
            const int e   = eidx[o0 + i];
            const float a = __expf(logits[e] - mx) * inv * ew[e];
            const float4 xj = ((const float4*)(xl + (long)src[e] * 128))[lane];
            acc.x = fmaf(xj.x, a, acc.x);
            acc.y = fmaf(xj.y, a, acc.y);
            acc.z = fmaf(xj.z, a, acc.z);
            acc.w = fmaf(xj.w, a, acc.w);
        }
    }
    ((float4*)(outp + (long)d * 128))[lane] = acc;
}

// ---------------------------------------------------------------------------
extern "C" void kernel_launch(void* const* d_in, const int* in_sizes, int n_in,
                              void* d_out, int out_size, void* d_ws,
                              size_t ws_size, hipStream_t stream) {
    (void)n_in; (void)out_size; (void)ws_size;

    const float* x_src = (const float*)d_in[0];
    const float* x_dst = (const float*)d_in[1];
    const float* eattr = (const float*)d_in[2];
    const float* ew    = (const float*)d_in[3];
    const float* W_l   = (const float*)d_in[4];
    const float* b_l   = (const float*)d_in[5];
    const float* W_r   = (const float*)d_in[6];
    const float* b_r   = (const float*)d_in[7];
    const float* att   = (const float*)d_in[8];
    const float* W_e   = (const float*)d_in[9];
    const float* b_e   = (const float*)d_in[10];
    const float* W1    = (const float*)d_in[11];
    const float* b1    = (const float*)d_in[12];
    const float* W2    = (const float*)d_in[13];
    const float* b2    = (const float*)d_in[14];
    const int*   src   = (const int*)d_in[15];
    const int*   dst   = (const int*)d_in[16];

    const int NSRC = in_sizes[0] / 128;
    const int NDST = in_sizes[1] / 128;
    const int E    = in_sizes[3];

    // ---- workspace layout (256B-aligned slices; hbuf aliases xl) ----
    auto align_up = [](size_t v) { return (v + 255) & ~(size_t)255; };
    char* w = (char*)d_ws;
    float* xl     = (float*)w;  w += align_up((size_t)NSRC * 128 * 4);
    float* xr     = (float*)w;  w += align_up((size_t)NDST * 128 * 4);
    float* logits = (float*)w;  w += align_up((size_t)E * 4);
    int*   eidx   = (int*)w;    w += align_up((size_t)E * 4);
    int*   deg    = (int*)w;    w += align_up((size_t)NDST * 4);
    int*   cursor = (int*)w;    w += align_up((size_t)NDST * 4);
    int*   off    = (int*)w;    w += align_up((size_t)(NDST + 1) * 4);
    float* outp   = (float*)w;  w += align_up((size_t)NDST * 128 * 4);
    float* hbuf   = xl;  // x_l is dead after dst_reduce
    float* out    = (float*)d_out;

    // 1) node projections (f32 WMMA)
    gemm128_wmma<0><<<NSRC / 16, 256, 0, stream>>>(x_src, W_l, b_l, xl, NSRC);
    gemm128_wmma<0><<<NDST / 16, 256, 0, stream>>>(x_dst, W_r, b_r, xr, NDST);

    // 2) CSR build (counting sort over dst)
    zero_counts_kernel<<<(NDST + 255) / 256, 256, 0, stream>>>(deg, cursor, NDST);
    hist_kernel<<<(E + 255) / 256, 256, 0, stream>>>(dst, deg, E);
    exscan_kernel<<<1, 1024, 0, stream>>>(deg, off, NDST);
    scatter_kernel<<<(E + 255) / 256, 256, 0, stream>>>(dst, off, cursor, eidx, E);

    // 3) edge logits (fused edge_attr @ W_e)
    edge_logits_kernel<<<(E + 7) / 8, 256, 0, stream>>>(
        xl, xr, eattr, W_e, b_e, att, src, dst, logits, E);

    // 4) segment softmax + gather-accumulate (+ residual)
    dst_reduce_kernel<<<(NDST + 7) / 8, 256, 0, stream>>>(
        xl, xr, logits, ew, off, eidx, src, outp, NDST);

    // 5) MLP (f32 WMMA): h = relu(outp@W1+b1); out = h@W2+b2
    gemm128_wmma<1><<<NDST / 16, 256, 0, stream>>>(outp, W1, b1, hbuf, NDST);
    gemm128_wmma<0><<<NDST / 16, 256, 0, stream>>>(hbuf, W2, b2, out, NDST);
}